// EventLossFunction_62586263437487
// MI455X (gfx1250) — compile-verified
//
#include <hip/hip_runtime.h>
#include <hip/hip_bf16.h>

// CDNA5 (gfx1250) fused kernel for the pairwise-BCE "event loss":
//   P = S @ S^T  (S: 12000x64 f32, rows sum to 1)
//   loss = -sum over kept pairs of [ y*log(clip(P)) + (1-y)*log(clip(1-P)) ] / (n*(n-1))
// Masked (row%1024==col) entries contribute exactly 0 in the reference, so we
// just skip them. The GEMM is done with the native f32 WMMA
// (v_wmma_f32_16x16x4_f32), K=64 as 16 chained K=4 steps, so no precision is
// lost versus the f32 reference before the logs.

typedef __attribute__((ext_vector_type(2))) float v2f;
typedef __attribute__((ext_vector_type(8))) float v8f;

#define N_NODES   12000
#define N_EDGES   64
#define BATCHSZ   1024
#define TILES     750        // 12000 / 16, exact
#define WPB       4          // waves per block (128 threads, wave32)

__global__ void zero_out_kernel(float* out) { out[0] = 0.0f; }

__global__ __launch_bounds__(128) void event_loss_wmma_kernel(
    const float* __restrict__ S,      // [N_NODES, N_EDGES] row-major f32
    const int*   __restrict__ labels, // [N_NODES]
    float*       __restrict__ out)    // [1] accumulated scalar
{
    __shared__ float wave_acc[WPB];

    const int lane = threadIdx.x & 31;
    const int wave = threadIdx.x >> 5;

    // One 16x16 tile of P per wave; grid covers all 750*750 tiles exactly.
    const long long t  = (long long)blockIdx.x * WPB + wave;
    const int ti = (int)(t / TILES);
    const int tj = (int)(t % TILES);
    const int i0 = ti * 16;
    const int j0 = tj * 16;

    const int r16  = lane & 15;   // row within the 16-row strip this lane loads
    const int half = lane >> 4;   // 0: K pair {0,1}; 1: K pair {2,3} (ISA A/B layout)

    // Per ISA 7.12.2, the 32-bit A (16x4) operand: lanes 0-15 hold M=lane with
    // VGPR0/1 = K0/K1; lanes 16-31 hold K2/K3. The 4x16 B operand mirrors this
    // with N striped across lanes. Both collapse to the same addressing on the
    // row-major S (B uses the column-tile's rows since B = S^T tile):
    const float* pa = S + (i0 + r16) * N_EDGES + half * 2;
    const float* pb = S + (j0 + r16) * N_EDGES + half * 2;

    v8f c = {};  // f32 accumulator, D=A*B+C chained across K
    #pragma unroll
    for (int k = 0; k < N_EDGES / 4; ++k) {
        v2f a = *(const v2f*)(pa + 4 * k);   // global_load_b64
        v2f b = *(const v2f*)(pb + 4 * k);
        // 8 args: (neg_a, A, neg_b, B, c_mod, C, reuse_a, reuse_b)
        c = __builtin_amdgcn_wmma_f32_16x16x4_f32(
                false, a, false, b, (short)0, c, false, false);
    }

    // C/D layout: lane l, vgpr r -> P[i0 + r + 8*(l>=16)][j0 + l%16]
    const int   gj = j0 + r16;
    const int   lj = labels[gj];

    float acc = 0.0f;
    #pragma unroll
    for (int r = 0; r < 8; ++r) {
        const int   gi   = i0 + r + 8 * half;
        const int   li   = labels[gi];
        const float pv   = c[r];
        const float p    = fminf(fmaxf(pv,        1e-38f), 1.0f);
        const float q    = fminf(fmaxf(1.0f - pv, 1e-38f), 1.0f);
        const float keep = ((gi % BATCHSZ) == gj) ? 0.0f : 1.0f;
        const float term = (li == lj) ? -__logf(p) : -__logf(q);
        acc += keep * term;
    }

    // wave32 butterfly reduction
    #pragma unroll
    for (int off = 16; off > 0; off >>= 1)
        acc += __shfl_xor(acc, off, 32);

    if (lane == 0) wave_acc[wave] = acc;
    __syncthreads();

    if (threadIdx.x == 0) {
        float s = 0.0f;
        #pragma unroll
        for (int w = 0; w < WPB; ++w) s += wave_acc[w];
        const float inv_pairs =
            1.0f / ((float)N_NODES * (float)(N_NODES - 1));
        atomicAdd(out, s * inv_pairs);
    }
}

extern "C" void kernel_launch(void* const* d_in, const int* in_sizes, int n_in,
                              void* d_out, int out_size, void* d_ws, size_t ws_size,
                              hipStream_t stream) {
    const float* S      = (const float*)d_in[0];  // softmax_output, f32 [12000,64]
    const int*   labels = (const int*)d_in[1];    // labels (int)    [12000]
    float*       out    = (float*)d_out;          // scalar f32

    zero_out_kernel<<<1, 1, 0, stream>>>(out);

    const int n_tiles = TILES * TILES;            // 562500, divisible by WPB
    dim3 grid(n_tiles / WPB);                     // 140625 blocks
    dim3 block(WPB * 32);                         // 128 threads = 4 waves
    event_loss_wmma_kernel<<<grid, block, 0, stream>>>(S, labels, out);
}